// HGTConv_s_13975823582060
// MI455X (gfx1250) — compile-verified
//
#include <hip/hip_runtime.h>
#include <math.h>

// HGT conv: N nodes, E edges, IN = H*D = 128, NT=3 node types, ET=4 edge types, fp32.
#define IN_DIM 128
#define HEADS  4
#define DHEAD  32
#define HD     128   // HEADS*DHEAD
#define NTYPES 3
#define ETYPES 4

typedef float v2f __attribute__((ext_vector_type(2)));
typedef float v8f __attribute__((ext_vector_type(8)));

// D = A(16x4,f32) * B(4x16,f32) + C(16x16,f32)  -> v_wmma_f32_16x16x4_f32
__device__ __forceinline__ v8f wmma4(v2f a, v2f b, v8f c) {
  return __builtin_amdgcn_wmma_f32_16x16x4_f32(false, a, false, b, (short)0, c,
                                               false, false);
}

// Async copy 16B global -> LDS per lane (CDNA5 GLOBAL_LOAD_ASYNC_TO_LDS_B128,
// tracked with ASYNCcnt; no VGPR staging). lds_addr = low 32 bits of flat
// pointer to __shared__ (== wave-relative LDS address per ISA 10.2).
__device__ __forceinline__ void async_copy16(void* lds_ptr, const void* gptr) {
  unsigned lds_addr = (unsigned)(unsigned long long)lds_ptr;
  asm volatile("global_load_async_to_lds_b128 %0, %1, off"
               :
               : "v"(lds_addr), "v"(gptr)
               : "memory");
}
__device__ __forceinline__ void async_wait() {
  asm volatile("s_wait_asynccnt 0x0" ::: "memory");
}

// sign-aware float atomic max via int/uint ordering trick
__device__ __forceinline__ void atomicMaxF(float* addr, float val) {
  if (val >= 0.0f)
    atomicMax((int*)addr, __float_as_int(val));
  else
    atomicMin((unsigned int*)addr, (unsigned int)__float_as_int(val));
}

__global__ void init_kernel(float* __restrict__ agg, float* __restrict__ amax,
                            float* __restrict__ esum, int n) {
  size_t i = (size_t)blockIdx.x * blockDim.x + threadIdx.x;
  size_t nAgg = (size_t)n * HD;
  size_t nH   = (size_t)n * HEADS;
  if (i < nAgg) agg[i] = 0.0f;
  if (i < nH) {
    amax[i] = -__builtin_huge_valf();
    esum[i] = 0.0f;
  }
}

// Y[r,:] = relu?( X[r,:] @ W[types[r]] )   X:[n,128] W:[T,128,128] Y:[n,128]
// One 16-row block per workgroup; 8 waves = 8 column tiles of 16.
// Per-type row masking: acc += mask_t(A) * B_t summed over t (exact typed linear).
template <bool RELU>
__global__ __launch_bounds__(256) void typed_linear_kernel(
    const float* __restrict__ X, const int* __restrict__ types,
    const float* __restrict__ W, float* __restrict__ Y, int n) {
  __shared__ float xls[16 * 132];  // padded stride: 132 floats (528B, 16B aligned)
  __shared__ int tls[16];
  const int tid = threadIdx.x;
  const int rowbase = blockIdx.x * 16;
  if (tid < 16) {
    int r = rowbase + tid;
    tls[tid] = (r < n) ? types[r] : -1;  // -1: row never matches any type
  }
  // stage X tile via async global->LDS (2 x b128 per thread); OOB rows clamp
  // their source index (data is neutralized by the type mask / store guards).
  for (int i = tid; i < 512; i += 256) {  // 16x128 floats as float4
    int row = i >> 5, c4 = i & 31;
    int r = rowbase + row;
    if (r >= n) r = n - 1;
    async_copy16(&xls[row * 132 + c4 * 4], &X[(size_t)r * HD + c4 * 4]);
  }
  async_wait();
  __syncthreads();
  const int lane = tid & 31;
  const int wv = tid >> 5;
  const int colbase = wv * 16;
  const int m = lane & 15;        // A row / C col index
  const int khalf = lane >> 4;    // 0: K lo pair, 1: K hi pair
  const int nn = lane & 15;       // B/C column
  v8f acc = {0.f, 0.f, 0.f, 0.f, 0.f, 0.f, 0.f, 0.f};
  const int tm = tls[m];
  for (int kk = 0; kk < IN_DIM; kk += 4) {
    int kb = kk + khalf * 2;
    float ax = xls[m * 132 + kb];
    float ay = xls[m * 132 + kb + 1];
#pragma unroll
    for (int t = 0; t < NTYPES; ++t) {
      v2f a;
      a.x = (tm == t) ? ax : 0.0f;
      a.y = (tm == t) ? ay : 0.0f;
      const float* Wp = W + ((size_t)t * IN_DIM + kb) * HD + colbase + nn;
      v2f b;
      b.x = Wp[0];     // B[kb][n]
      b.y = Wp[HD];    // B[kb+1][n]
      acc = wmma4(a, b, acc);
    }
  }
#pragma unroll
  for (int r = 0; r < 8; ++r) {
    int row = r + khalf * 8;
    int gr = rowbase + row;
    if (gr < n) {
      float v = acc[r];
      if (RELU) v = fmaxf(v, 0.0f);
      Y[(size_t)gr * HD + colbase + nn] = v;
    }
  }
}

// Edge pass. 16 edges per workgroup, 4 waves = 4 heads.
// kw[e,h,:] = kqv[src[e],h,:] @ rel_att[h, etype[e]]   (v_e == k_e, so m == kw)
// SCORE:  a = (kw . q) * rel_pri / sqrt(D); atomicMax segment max.
// !SCORE: recompute kw, scale by ex/esum[dst], atomic scatter-add to agg.
template <bool SCORE>
__global__ __launch_bounds__(128) void edge_kernel(
    const float* __restrict__ kqv, const int* __restrict__ src,
    const int* __restrict__ dstArr, const int* __restrict__ etype,
    const float* __restrict__ rel_att, const float* __restrict__ rel_pri,
    float* __restrict__ a_buf, float* __restrict__ amax,
    const float* __restrict__ esum, float* __restrict__ agg, int nEdges) {
  __shared__ float kls[16 * 132];
  __shared__ float qls[16 * 132];
  __shared__ int sL[16], dL[16], eL[16];
  const int tid = threadIdx.x;
  const int base = blockIdx.x * 16;
  if (tid < 16) {
    int e = base + tid;
    if (e >= nEdges) e = nEdges - 1;
    sL[tid] = src[e];
    dL[tid] = dstArr[e];
    eL[tid] = (base + tid < nEdges) ? etype[e] : -1;  // -1: masks A rows to 0
  }
  __syncthreads();
  // gather k (=v) and q rows straight into LDS: 2 bufs * 16 rows * 32 float4
  // = 1024 b128 async copies, 8 per thread, no VGPR staging.
  for (int i = tid; i < 1024; i += 128) {
    int buf = i >> 9;
    int rem = i & 511;
    int row = rem >> 5, c4 = rem & 31;
    int node = buf ? dL[row] : sL[row];
    float* lptr = (buf ? qls : kls) + row * 132 + c4 * 4;
    async_copy16(lptr, &kqv[(size_t)node * HD + c4 * 4]);
  }
  async_wait();
  __syncthreads();
  const int lane = tid & 31;
  const int h = tid >> 5;  // head per wave
  const int m = lane & 15;
  const int khalf = lane >> 4;
  const int nn = lane & 15;
  v8f acc0 = {0.f, 0.f, 0.f, 0.f, 0.f, 0.f, 0.f, 0.f};  // cols 0..15
  v8f acc1 = {0.f, 0.f, 0.f, 0.f, 0.f, 0.f, 0.f, 0.f};  // cols 16..31
  const int tm = eL[m];
#pragma unroll
  for (int t = 0; t < ETYPES; ++t) {
    const float* B = rel_att + ((size_t)h * ETYPES + t) * DHEAD * DHEAD;
#pragma unroll
    for (int kk = 0; kk < DHEAD; kk += 4) {
      int kb = kk + khalf * 2;
      bool on = (tm == t);
      v2f a;
      a.x = on ? kls[m * 132 + h * DHEAD + kb] : 0.0f;
      a.y = on ? kls[m * 132 + h * DHEAD + kb + 1] : 0.0f;
      const float* Bp = B + kb * DHEAD + nn;
      v2f b0, b1;
      b0.x = Bp[0];            // B[kb][n]
      b0.y = Bp[DHEAD];        // B[kb+1][n]
      b1.x = Bp[16];           // B[kb][n+16]
      b1.y = Bp[DHEAD + 16];   // B[kb+1][n+16]
      acc0 = wmma4(a, b0, acc0);
      acc1 = wmma4(a, b1, acc1);
    }
  }
  if (SCORE) {
#pragma unroll
    for (int r = 0; r < 8; ++r) {
      int row = r + khalf * 8;
      float q0 = qls[row * 132 + h * DHEAD + nn];
      float q1 = qls[row * 132 + h * DHEAD + 16 + nn];
      float p = acc0[r] * q0 + acc1[r] * q1;
      // reduce across the 16-lane group (wave32; xor 1,2,4,8 keeps halves intact)
      p += __shfl_xor(p, 1);
      p += __shfl_xor(p, 2);
      p += __shfl_xor(p, 4);
      p += __shfl_xor(p, 8);
      if (nn == 0) {
        int e = base + row;
        if (e < nEdges) {
          int t = eL[row];
          float av = p * rel_pri[h * ETYPES + t] * 0.17677669529663687f; // 1/sqrt(32)
          a_buf[(size_t)e * HEADS + h] = av;
          atomicMaxF(&amax[(size_t)dL[row] * HEADS + h], av);
        }
      }
    }
  } else {
#pragma unroll
    for (int r = 0; r < 8; ++r) {
      int row = r + khalf * 8;
      int e = base + row;
      if (e < nEdges) {
        int d = dL[row];
        float ex = a_buf[(size_t)e * HEADS + h];
        float attn = ex / esum[(size_t)d * HEADS + h];
        atomicAdd(&agg[(size_t)d * HD + h * DHEAD + nn], acc0[r] * attn);
        atomicAdd(&agg[(size_t)d * HD + h * DHEAD + 16 + nn], acc1[r] * attn);
      }
    }
  }
}

__global__ void softmax_norm_kernel(float* __restrict__ a_buf,
                                    const float* __restrict__ amax,
                                    float* __restrict__ esum,
                                    const int* __restrict__ dstArr, int nEdges) {
  size_t i = (size_t)blockIdx.x * blockDim.x + threadIdx.x;
  if (i >= (size_t)nEdges * HEADS) return;
  int e = (int)(i / HEADS), h = (int)(i % HEADS);
  int d = dstArr[e];
  float ex = expf(a_buf[i] - amax[(size_t)d * HEADS + h]);
  a_buf[i] = ex;
  atomicAdd(&esum[(size_t)d * HEADS + h], ex);
}

extern "C" void kernel_launch(void* const* d_in, const int* in_sizes, int n_in,
                              void* d_out, int out_size, void* d_ws, size_t ws_size,
                              hipStream_t stream) {
  const float* x       = (const float*)d_in[0];
  const int*   ntype   = (const int*)d_in[1];
  const int*   etype   = (const int*)d_in[2];
  const int*   src     = (const int*)d_in[3];
  const int*   dst     = (const int*)d_in[4];
  const float* W_kqv   = (const float*)d_in[5];
  const float* W_a     = (const float*)d_in[6];
  const float* rel_pri = (const float*)d_in[7];
  const float* rel_att = (const float*)d_in[8];
  const int n  = in_sizes[1];  // #nodes (ntype count)
  const int ne = in_sizes[2];  // #edges (etype count)

  float* ws    = (float*)d_ws;
  float* kqv   = ws;                            // n*128
  float* agg   = kqv + (size_t)n * HD;          // n*128
  float* a_buf = agg + (size_t)n * HD;          // ne*4
  float* amax  = a_buf + (size_t)ne * HEADS;    // n*4
  float* esum  = amax + (size_t)n * HEADS;      // n*4

  {
    size_t tot = (size_t)n * HD;
    init_kernel<<<(unsigned)((tot + 255) / 256), 256, 0, stream>>>(agg, amax, esum, n);
  }
  const int ntiles = (n + 15) / 16;
  const int etiles = (ne + 15) / 16;

  typed_linear_kernel<false><<<ntiles, 256, 0, stream>>>(x, ntype, W_kqv, kqv, n);

  edge_kernel<true><<<etiles, 128, 0, stream>>>(kqv, src, dst, etype, rel_att,
                                                rel_pri, a_buf, amax, nullptr,
                                                nullptr, ne);
  {
    size_t tot = (size_t)ne * HEADS;
    softmax_norm_kernel<<<(unsigned)((tot + 255) / 256), 256, 0, stream>>>(
        a_buf, amax, esum, dst, ne);
  }
  edge_kernel<false><<<etiles, 128, 0, stream>>>(kqv, src, dst, etype, rel_att,
                                                 rel_pri, a_buf, nullptr, esum,
                                                 agg, ne);

  typed_linear_kernel<true><<<ntiles, 256, 0, stream>>>(agg, ntype, W_a,
                                                        (float*)d_out, n);
}